// GNN_Net_77266461655141
// MI455X (gfx1250) — compile-verified
//
#include <hip/hip_runtime.h>

// GNN (5x GCNConv + BN + LeakyReLU, scatter-mean pool, MLP head) for MI455X.
// fp32 end-to-end; GEMMs use V_WMMA_F32_16X16X4_F32 (CDNA5 wave32 WMMA) with
// the weight matrix staged transposed in LDS (ds_load_b64 B-operand fetches).

#define NF 96          // padded node-feature stride (max channel width)
#define NGRAPH 256
#define BNEPS 1e-5f
#define SLOPE 0.01f

typedef __attribute__((ext_vector_type(2))) float v2f;
typedef __attribute__((ext_vector_type(8))) float v8f;

__device__ __forceinline__ float lrelu(float v) { return v >= 0.0f ? v : SLOPE * v; }

// ---------------- utility kernels ----------------
__global__ void zero_f(float* __restrict__ p, int n) {
  int i = blockIdx.x * blockDim.x + threadIdx.x;
  if (i < n) p[i] = 0.0f;
}

// copy x [N,3] into A with stride NF, pad K=3 (col 3) with zero for the K=4 WMMA step
__global__ void prep_x(const float* __restrict__ x, float* __restrict__ A, int N) {
  int i = blockIdx.x * blockDim.x + threadIdx.x;
  if (i >= N) return;
  float* r = A + (size_t)i * NF;
  r[0] = x[3 * i + 0];
  r[1] = x[3 * i + 1];
  r[2] = x[3 * i + 2];
  r[3] = 0.0f;
}

__global__ void deg_accum(const int* __restrict__ col, float* __restrict__ deg, int nE) {
  int e = blockIdx.x * blockDim.x + threadIdx.x;
  if (e < nE) atomicAdd(&deg[col[e]], 1.0f);
}

// deg -> dinv in place; +1 accounts for the self loop edge
__global__ void dinv_k(float* __restrict__ d, int N) {
  int i = blockIdx.x * blockDim.x + threadIdx.x;
  if (i < N) d[i] = rsqrtf(fmaxf(d[i] + 1.0f, 1.0f));
}

// ---------------- WMMA GEMM: H[N,Fo] = A[N,Kd(<=96,stride NF)] @ W[Kd,Fo] ----------------
// one wave per 16-row strip; NT = Fo/16 output tiles held in registers.
// W is staged in LDS transposed to [Fo][Kd] so each B fetch is one ds_load_b64.
template <int NT>
__global__ __launch_bounds__(256) void gcn_gemm_wmma(const float* __restrict__ X,
                                                     const float* __restrict__ W,
                                                     float* __restrict__ H,
                                                     int nTilesM, int Kdim, int Fo) {
  __shared__ float sWt[96 * 64];  // [Fo][Kd] transposed weights, max 24 KB
  int tid = threadIdx.x;
  int total = Kdim * Fo;
  for (int i = tid; i < total; i += blockDim.x) {
    int k = i / Fo, n = i % Fo;
    sWt[n * Kdim + k] = W[i];
  }
  __syncthreads();

  int wave = tid >> 5;
  int lane = tid & 31;
  int tileM = blockIdx.x * (blockDim.x >> 5) + wave;
  if (tileM >= nTilesM) return;  // wave-uniform: EXEC stays all-ones for WMMA
  int m0 = tileM << 4;
  int half = lane >> 4;          // 0: K pair (k0,k0+1)  1: K pair (k0+2,k0+3)
  int l = lane & 15;

  v8f acc[NT] = {};
  // A operand: row m0+l, K pair starting at k0 + 2*half  (8B aligned -> b64 load)
  const float* xr = X + (size_t)(m0 + l) * NF + 2 * half;
  const float* wl = sWt + (size_t)l * Kdim + 2 * half;
  for (int k0 = 0; k0 < Kdim; k0 += 4) {
    v2f a = *(const v2f*)(xr + k0);
#pragma unroll
    for (int t = 0; t < NT; ++t) {
      v2f b = *(const v2f*)(wl + (size_t)t * 16 * Kdim + k0);  // ds_load_b64
      acc[t] = __builtin_amdgcn_wmma_f32_16x16x4_f32(false, a, false, b,
                                                     (short)0, acc[t], false, false);
    }
  }
  // C/D layout: VGPR r -> rows m0+r (lanes 0-15) and m0+r+8 (lanes 16-31), N = l
#pragma unroll
  for (int t = 0; t < NT; ++t) {
#pragma unroll
    for (int r = 0; r < 8; ++r) {
      H[(size_t)(m0 + r + 8 * half) * NF + t * 16 + l] = acc[t][r];
    }
  }
}

// Y[i,f] = bias[f] + dinv[i]^2 * H[i,f]   (self loop contribution + bias)
__global__ void init_out(const float* __restrict__ H, const float* __restrict__ dinv,
                         const float* __restrict__ bias, float* __restrict__ Y,
                         int N, int Fo) {
  int idx = blockIdx.x * blockDim.x + threadIdx.x;
  if (idx >= N * Fo) return;
  int i = idx / Fo, f = idx % Fo;
  float d = dinv[i];
  Y[(size_t)i * NF + f] = bias[f] + d * d * H[(size_t)i * NF + f];
}

// one wave per edge: lanes cover channels -> coalesced gather + coalesced float atomics
__global__ void edge_scatter(const int* __restrict__ row, const int* __restrict__ colv,
                             const float* __restrict__ dinv, const float* __restrict__ H,
                             float* __restrict__ Y, int nE, int Fo) {
  int gwave = (blockIdx.x * blockDim.x + threadIdx.x) >> 5;
  int lane = threadIdx.x & 31;
  int nw = (gridDim.x * blockDim.x) >> 5;
  for (int e = gwave; e < nE; e += nw) {
    if (e + 4096 < nE) {  // speculative prefetch of upcoming edge indices
      __builtin_prefetch(&row[e + 4096], 0, 0);
      __builtin_prefetch(&colv[e + 4096], 0, 0);
    }
    int r = row[e], c = colv[e];
    float nrm = dinv[r] * dinv[c];
    const float* hr = H + (size_t)r * NF;
    float* yc = Y + (size_t)c * NF;
    for (int f = lane; f < Fo; f += 32) atomicAdd(&yc[f], nrm * hr[f]);
  }
}

// per-block LDS channel reduction, then global atomics into stats[0..Fo)=sum, [NF..NF+Fo)=sumsq
__global__ void bn_stats(const float* __restrict__ Y, float* __restrict__ stats,
                         int N, int Fo) {
  __shared__ float s[2 * NF];
  for (int i = threadIdx.x; i < 2 * NF; i += blockDim.x) s[i] = 0.0f;
  __syncthreads();
  const int CH = 128;  // nodes per block
  int start = blockIdx.x * CH;
  int cntn = N - start;
  if (cntn > CH) cntn = CH;
  if (cntn > 0) {
    int total = cntn * Fo;
    for (int idx = threadIdx.x; idx < total; idx += blockDim.x) {
      int i = start + idx / Fo, f = idx % Fo;
      float v = Y[(size_t)i * NF + f];
      atomicAdd(&s[f], v);
      atomicAdd(&s[NF + f], v * v);
    }
  }
  __syncthreads();
  for (int f = threadIdx.x; f < Fo; f += blockDim.x) {
    atomicAdd(&stats[f], s[f]);
    atomicAdd(&stats[NF + f], s[NF + f]);
  }
}

// ss[f]=scale, ss[NF+f]=shift   (biased variance, training-mode BN)
__global__ void bn_finalize(const float* __restrict__ stats, const float* __restrict__ g,
                            const float* __restrict__ b, float* __restrict__ ss,
                            float invN, int Fo) {
  int f = threadIdx.x;
  if (f >= Fo) return;
  float mean = stats[f] * invN;
  float var = stats[NF + f] * invN - mean * mean;
  float sc = g[f] * rsqrtf(var + BNEPS);
  ss[f] = sc;
  ss[NF + f] = b[f] - mean * sc;
}

__global__ void bn_apply(float* __restrict__ Y, const float* __restrict__ ss, int N, int Fo) {
  int idx = blockIdx.x * blockDim.x + threadIdx.x;
  if (idx >= N * Fo) return;
  int i = idx / Fo, f = idx % Fo;
  size_t o = (size_t)i * NF + f;
  Y[o] = lrelu(Y[o] * ss[f] + ss[NF + f]);
}

// ---------------- pooling ----------------
__global__ void pool_cnt_k(const int* __restrict__ batch, float* __restrict__ cnt, int N) {
  int i = blockIdx.x * blockDim.x + threadIdx.x;
  if (i < N) atomicAdd(&cnt[batch[i]], 1.0f);
}
__global__ void pool_sum_k(const float* __restrict__ X, const int* __restrict__ batch,
                           float* __restrict__ pool, int N) {
  int idx = blockIdx.x * blockDim.x + threadIdx.x;
  if (idx >= N * NF) return;
  int i = idx / NF, f = idx % NF;
  atomicAdd(&pool[batch[i] * NF + f], X[(size_t)i * NF + f]);
}
__global__ void pool_div_k(const float* __restrict__ pool, const float* __restrict__ cnt,
                           float* __restrict__ g, int n) {
  int idx = blockIdx.x * blockDim.x + threadIdx.x;
  if (idx < n) g[idx] = pool[idx] / fmaxf(cnt[idx / NF], 1.0f);
}

// ---------------- tiny head (256 rows) ----------------
__global__ void fc_k(const float* __restrict__ in, const float* __restrict__ W,
                     const float* __restrict__ b, float* __restrict__ out,
                     int R, int Fi, int Fo) {
  int idx = blockIdx.x * blockDim.x + threadIdx.x;
  if (idx >= R * Fo) return;
  int r = idx / Fo, c = idx % Fo;
  float acc = b[c];
  const float* ir = in + (size_t)r * Fi;
  for (int k = 0; k < Fi; ++k) acc += ir[k] * W[k * Fo + c];
  out[idx] = acc;
}

// BN over R rows (one thread per channel), then LeakyReLU, in place
__global__ void bn_head_k(float* __restrict__ x, const float* __restrict__ g,
                          const float* __restrict__ b, int R, int Fo) {
  int c = blockIdx.x * blockDim.x + threadIdx.x;
  if (c >= Fo) return;
  float s = 0.0f, s2 = 0.0f;
  for (int r = 0; r < R; ++r) {
    float v = x[r * Fo + c];
    s += v;
    s2 += v * v;
  }
  float mean = s / R;
  float var = s2 / R - mean * mean;
  float sc = g[c] * rsqrtf(var + BNEPS);
  float sh = b[c] - mean * sc;
  for (int r = 0; r < R; ++r) x[r * Fo + c] = lrelu(x[r * Fo + c] * sc + sh);
}

__global__ void copy_k(const float* __restrict__ src, float* __restrict__ dst, int n) {
  int i = blockIdx.x * blockDim.x + threadIdx.x;
  if (i < n) dst[i] = src[i];
}

// ---------------- launcher ----------------
extern "C" void kernel_launch(void* const* d_in, const int* in_sizes, int n_in,
                              void* d_out, int out_size, void* d_ws, size_t ws_size,
                              hipStream_t stream) {
  (void)n_in; (void)out_size; (void)ws_size;
  const float* x   = (const float*)d_in[0];
  const int* eidx  = (const int*)d_in[1];
  const int* batch = (const int*)d_in[2];
  const float* P[34];
  for (int i = 0; i < 34; ++i) P[i] = (const float*)d_in[3 + i];

  const int N = in_sizes[0] / 3;   // 100000 (multiple of 16)
  const int nE = in_sizes[1] / 2;  // 1600000
  const int* row = eidx;           // edge_index[0] = sources
  const int* col = eidx + nE;      // edge_index[1] = targets

  float* ws = (float*)d_ws;
  float* A = ws;                       // [N, NF] activations / aggregation target
  float* H = A + (size_t)N * NF;       // [N, NF] GEMM output (messages)
  float* dinv = H + (size_t)N * NF;    // [N]
  float* stats = dinv + N;             // [2*NF]
  float* ss = stats + 2 * NF;          // [2*NF]
  float* pool = ss + 2 * NF;           // [NGRAPH*NF]
  float* cnt = pool + NGRAPH * NF;     // [NGRAPH]
  float* gbuf = cnt + NGRAPH;          // [NGRAPH*96]
  float* h1 = gbuf + NGRAPH * NF;      // [NGRAPH*128]
  float* feat = h1 + NGRAPH * 128;     // [NGRAPH*96]

  const int T = 256;
  auto cdiv = [](int a, int b) { return (a + b - 1) / b; };

  // degrees -> dinv
  zero_f<<<cdiv(N, T), T, 0, stream>>>(dinv, N);
  deg_accum<<<cdiv(nE, T), T, 0, stream>>>(col, dinv, nE);
  dinv_k<<<cdiv(N, T), T, 0, stream>>>(dinv, N);

  prep_x<<<cdiv(N, T), T, 0, stream>>>(x, A, N);

  const int Kds[5] = {4, 16, 32, 48, 64};   // layer-1 K padded 3->4 (col 3 zeroed)
  const int Fos[5] = {16, 32, 48, 64, 96};
  const int tilesM = N / 16;
  const int gemmBlocks = cdiv(tilesM, T / 32);

  for (int L = 0; L < 5; ++L) {
    int Kd = Kds[L], Fo = Fos[L];
    const float* W = P[4 * L + 0];
    const float* b = P[4 * L + 1];
    const float* bg = P[4 * L + 2];
    const float* bb = P[4 * L + 3];
    switch (Fo >> 4) {
      case 1: gcn_gemm_wmma<1><<<gemmBlocks, T, 0, stream>>>(A, W, H, tilesM, Kd, Fo); break;
      case 2: gcn_gemm_wmma<2><<<gemmBlocks, T, 0, stream>>>(A, W, H, tilesM, Kd, Fo); break;
      case 3: gcn_gemm_wmma<3><<<gemmBlocks, T, 0, stream>>>(A, W, H, tilesM, Kd, Fo); break;
      case 4: gcn_gemm_wmma<4><<<gemmBlocks, T, 0, stream>>>(A, W, H, tilesM, Kd, Fo); break;
      default: gcn_gemm_wmma<6><<<gemmBlocks, T, 0, stream>>>(A, W, H, tilesM, Kd, Fo); break;
    }
    init_out<<<cdiv(N * Fo, T), T, 0, stream>>>(H, dinv, b, A, N, Fo);
    edge_scatter<<<2048, T, 0, stream>>>(row, col, dinv, H, A, nE, Fo);
    zero_f<<<1, 2 * NF, 0, stream>>>(stats, 2 * NF);
    bn_stats<<<cdiv(N, 128), T, 0, stream>>>(A, stats, N, Fo);
    bn_finalize<<<1, 128, 0, stream>>>(stats, bg, bb, ss, 1.0f / (float)N, Fo);
    bn_apply<<<cdiv(N * Fo, T), T, 0, stream>>>(A, ss, N, Fo);
  }

  // scatter-mean pooling over graphs
  zero_f<<<cdiv(NGRAPH * NF + NGRAPH, T), T, 0, stream>>>(pool, NGRAPH * NF + NGRAPH);
  pool_cnt_k<<<cdiv(N, T), T, 0, stream>>>(batch, cnt, N);
  pool_sum_k<<<cdiv(N * NF, T), T, 0, stream>>>(A, batch, pool, N);
  pool_div_k<<<cdiv(NGRAPH * NF, T), T, 0, stream>>>(pool, cnt, gbuf, NGRAPH * NF);

  // MLP head: fc1(96->128)+BN+lrelu, fc2(128->96)+BN+lrelu
  fc_k<<<cdiv(NGRAPH * 128, T), T, 0, stream>>>(gbuf, P[20], P[21], h1, NGRAPH, 96, 128);
  bn_head_k<<<1, 128, 0, stream>>>(h1, P[22], P[23], NGRAPH, 128);
  fc_k<<<cdiv(NGRAPH * 96, T), T, 0, stream>>>(h1, P[24], P[25], feat, NGRAPH, 128, 96);
  bn_head_k<<<1, 96, 0, stream>>>(feat, P[26], P[27], NGRAPH, 96);

  // outputs: [pose(256x2) | depth(256x1) | logits(256x10) | feat(256x96)]
  float* out = (float*)d_out;
  fc_k<<<cdiv(NGRAPH * 2, T), T, 0, stream>>>(feat, P[28], P[29], out, NGRAPH, 96, 2);
  fc_k<<<cdiv(NGRAPH * 1, T), T, 0, stream>>>(feat, P[30], P[31], out + NGRAPH * 2, NGRAPH, 96, 1);
  fc_k<<<cdiv(NGRAPH * 10, T), T, 0, stream>>>(feat, P[32], P[33], out + NGRAPH * 3, NGRAPH, 96, 10);
  copy_k<<<cdiv(NGRAPH * 96, T), T, 0, stream>>>(feat, out + NGRAPH * 13, NGRAPH * 96);
}